// pointconv_19318762897744
// MI455X (gfx1250) — compile-verified
//
#include <hip/hip_runtime.h>
#include <hip/hip_bf16.h>
#include <math.h>

typedef float v2f __attribute__((ext_vector_type(2)));
typedef float v8f __attribute__((ext_vector_type(8)));

#define DD   64      // feature dim
#define KNN  24      // neighbors
#define NPTS 1024    // points per batch (problem-fixed)

// ---------------------------------------------------------------------------
// CDNA5 async global->LDS copies (ASYNCcnt-tracked). lds_off is the byte
// offset within the workgroup's LDS allocation (static __shared__ starts at 0).
// ---------------------------------------------------------------------------
__device__ __forceinline__ void async_copy_b128(unsigned lds_off, const void* g) {
  asm volatile("global_load_async_to_lds_b128 %0, %1, off"
               :: "v"(lds_off), "v"((unsigned long long)g) : "memory");
}
__device__ __forceinline__ void async_copy_b32(unsigned lds_off, const void* g) {
  asm volatile("global_load_async_to_lds_b32 %0, %1, off"
               :: "v"(lds_off), "v"((unsigned long long)g) : "memory");
}
__device__ __forceinline__ void async_wait0() {
  asm volatile("s_wait_asynccnt 0x0" ::: "memory");
}

// ---------------------------------------------------------------------------
// Wave-level 16x64 GEMM slab using V_WMMA_F32_16X16X4_F32 (full f32 precision).
// A: 16 rows x 64 cols (row-major, stride lda; global or LDS).
// W: 64x64 row-major (LDS-resident).
// acc[t] accumulates the 16x16 tile for columns [t*16, t*16+16).
// ---------------------------------------------------------------------------
__device__ __forceinline__ void wmma_gemm_16x64(const float* __restrict__ Ablk,
                                                int lda,
                                                const float* __restrict__ W,
                                                v8f acc[4]) {
  const int lane = threadIdx.x & 31;
  const int row  = lane & 15;
  const int kh   = (lane >> 4) << 1;   // 0 or 2
  for (int kb = 0; kb < DD; kb += 4) {
    v2f a;
    a.x = Ablk[row * lda + kb + kh + 0];
    a.y = Ablk[row * lda + kb + kh + 1];
#pragma unroll
    for (int t = 0; t < 4; ++t) {
      v2f b;
      b.x = W[(kb + kh + 0) * DD + t * 16 + row];
      b.y = W[(kb + kh + 1) * DD + t * 16 + row];
      acc[t] = __builtin_amdgcn_wmma_f32_16x16x4_f32(
          /*neg_a=*/false, a, /*neg_b=*/false, b,
          /*c_mod=*/(short)0, acc[t], /*reuse_a=*/false, /*reuse_b=*/false);
    }
  }
}

// ---------------------------------------------------------------------------
// feat = relu(relu(X@W1 + b1) @ W2 + b2)      X:[M,64]  W1,W2:[64,64]
// 128 threads = 4 waves; each wave owns a 16-row slab. Requires M % 64 == 0.
// W1/W2 staged into LDS via async-to-LDS b128; hidden activation staged
// through LDS to convert WMMA C-layout -> A-layout for the second layer.
// LDS: 16KB (w1) + 16KB (w2) + 16KB (hid) = 48KB/block.
// ---------------------------------------------------------------------------
__global__ __launch_bounds__(128)
void mlp2_kernel(const float* __restrict__ X,
                 const float* __restrict__ W1, const float* __restrict__ B1,
                 const float* __restrict__ W2, const float* __restrict__ B2,
                 float* __restrict__ Y, int M) {
  __shared__ __align__(16) float sbuf[3 * DD * DD];   // w1 | w2 | hid
  float* w1l = sbuf;
  float* w2l = sbuf + DD * DD;
  float* hid = sbuf + 2 * DD * DD;

  const int tid  = threadIdx.x;
  const int wave = tid >> 5;
  const int lane = tid & 31;
  const int row  = lane & 15;
  const int hi   = lane >> 4;
  const int rowbase = (blockIdx.x * 4 + wave) * 16;

  // stage both weight matrices into LDS (async, 16B per lane-op)
#pragma unroll
  for (int j = 0; j < (DD * DD) / (128 * 4); ++j) {     // 8 iters
    const int e4 = tid + j * 128;                        // float4 index
    async_copy_b128((unsigned)(e4 * 16), W1 + e4 * 4);
    async_copy_b128((unsigned)(DD * DD * 4 + e4 * 16), W2 + e4 * 4);
  }
  async_wait0();
  __syncthreads();

  v8f acc[4];
#pragma unroll
  for (int t = 0; t < 4; ++t)
    acc[t] = (v8f){0.f, 0.f, 0.f, 0.f, 0.f, 0.f, 0.f, 0.f};

  // layer 1: A from global, B from LDS
  wmma_gemm_16x64(X + (size_t)rowbase * DD, DD, w1l, acc);

  // bias + relu, C-layout -> LDS (row-major slab, per-wave region)
  float* hslab = hid + wave * 16 * DD;
#pragma unroll
  for (int t = 0; t < 4; ++t) {
    const int n   = t * 16 + row;
    const float bv = B1[n];
#pragma unroll
    for (int r = 0; r < 8; ++r) {
      const int m = r + hi * 8;
      hslab[m * DD + n] = fmaxf(acc[t][r] + bv, 0.f);
    }
  }
  __syncthreads();

  // layer 2: A and B both from LDS
#pragma unroll
  for (int t = 0; t < 4; ++t)
    acc[t] = (v8f){0.f, 0.f, 0.f, 0.f, 0.f, 0.f, 0.f, 0.f};
  wmma_gemm_16x64(hslab, DD, w2l, acc);

#pragma unroll
  for (int t = 0; t < 4; ++t) {
    const int n   = t * 16 + row;
    const float bv = B2[n];
#pragma unroll
    for (int r = 0; r < 8; ++r) {
      const int m = r + hi * 8;
      Y[(size_t)(rowbase + m) * DD + n] = fmaxf(acc[t][r] + bv, 0.f);
    }
  }
}

// ---------------------------------------------------------------------------
// Y = X @ W + b  (no relu). Same slab structure; W staged to LDS (16KB).
// ---------------------------------------------------------------------------
__global__ __launch_bounds__(128)
void gemm_bias_kernel(const float* __restrict__ X,
                      const float* __restrict__ W, const float* __restrict__ B,
                      float* __restrict__ Y, int M) {
  __shared__ __align__(16) float wl[DD * DD];
  const int tid  = threadIdx.x;
  const int wave = tid >> 5;
  const int lane = tid & 31;
  const int row  = lane & 15;
  const int hi   = lane >> 4;
  const int rowbase = (blockIdx.x * 4 + wave) * 16;

#pragma unroll
  for (int j = 0; j < (DD * DD) / (128 * 4); ++j) {
    const int e4 = tid + j * 128;
    async_copy_b128((unsigned)(e4 * 16), W + e4 * 4);
  }
  async_wait0();
  __syncthreads();

  v8f acc[4];
#pragma unroll
  for (int t = 0; t < 4; ++t)
    acc[t] = (v8f){0.f, 0.f, 0.f, 0.f, 0.f, 0.f, 0.f, 0.f};
  wmma_gemm_16x64(X + (size_t)rowbase * DD, DD, wl, acc);
#pragma unroll
  for (int t = 0; t < 4; ++t) {
    const int n   = t * 16 + row;
    const float bv = B[n];
#pragma unroll
    for (int r = 0; r < 8; ++r) {
      const int m = r + hi * 8;
      Y[(size_t)(rowbase + m) * DD + n] = acc[t][r] + bv;
    }
  }
}

// ---------------------------------------------------------------------------
// Brute-force KNN: one wave per query point; dist table in LDS, K iterative
// min-selection passes with wave32 butterfly reductions.
// idxOut[q*K + k] = per-batch index of k-th nearest (self included, like ref).
// ---------------------------------------------------------------------------
__global__ __launch_bounds__(256)
void knn_kernel(const float* __restrict__ xyz, int* __restrict__ idxOut, int N) {
  __shared__ float dist[8][NPTS];             // 32 KB
  const int wave = threadIdx.x >> 5;
  const int lane = threadIdx.x & 31;
  const int q = blockIdx.x * 8 + wave;        // global point id in [0, B*N)
  const int b = q / N;
  const int n = q % N;
  const float* xb = xyz + (size_t)b * N * 3;
  const float qx = xb[n * 3 + 0], qy = xb[n * 3 + 1], qz = xb[n * 3 + 2];

  for (int c = lane; c < N; c += 32) {
    const float dx = xb[c * 3 + 0] - qx;
    const float dy = xb[c * 3 + 1] - qy;
    const float dz = xb[c * 3 + 2] - qz;
    dist[wave][c] = dx * dx + dy * dy + dz * dz;
  }

  for (int kk = 0; kk < KNN; ++kk) {
    float best = 3.4e38f;
    int   bi   = 0x7fffffff;
    for (int c = lane; c < N; c += 32) {
      const float v = dist[wave][c];
      if (v < best || (v == best && c < bi)) { best = v; bi = c; }
    }
#pragma unroll
    for (int off = 16; off > 0; off >>= 1) {
      const float ov = __shfl_xor(best, off, 32);
      const int   oi = __shfl_xor(bi, off, 32);
      if (ov < best || (ov == best && oi < bi)) { best = ov; bi = oi; }
    }
    if (lane == 0) idxOut[(size_t)q * KNN + kk] = bi;
    dist[wave][bi] = 3.4e38f;                 // bi uniform after reduction
  }
}

// ---------------------------------------------------------------------------
// Fused edge-weight generation + softmax(axis=K*D) + weighted aggregation.
// One 64-thread block per point; thread = output column d2. Online softmax
// over m = (k,d1) in 0..1535 avoids materializing w (805 MB in the reference).
// Neighbor features gathered into LDS via async-to-LDS b128 copies; rw/rb
// warmed into cache with global_prefetch, then re-read from WGP$/L2.
// ---------------------------------------------------------------------------
__global__ __launch_bounds__(64)
void agg_kernel(const float* __restrict__ xyz, const float* __restrict__ feat,
                const int* __restrict__ idx, const float* __restrict__ rw,
                const float* __restrict__ rb, float* __restrict__ out, int N) {
  // LDS layout (single buffer, base offset 0):
  //   [0, KNN*DD)                 nf   neighbor features
  //   [KNN*DD, KNN*DD+KNN*4)      rel  (padded to 4)
  //   [KNN*DD+KNN*4, ... +KNN)    nbr  (int)
  __shared__ __align__(16) float sbuf[KNN * DD + KNN * 4 + KNN];
  float* nf   = sbuf;
  float* relS = sbuf + KNN * DD;
  int*   nbrS = (int*)(sbuf + KNN * DD + KNN * 4);

  const int tid = threadIdx.x;                // d2 in [0,64)
  const int q = blockIdx.x;
  const int b = q / N, n = q % N;

  // warm rw (48KB) + rb (16KB) into cache, 128B lines per prefetch
  for (int t = tid; t < 384; t += 64)
    __builtin_prefetch((const char*)rw + t * 128, 0, 3);
  for (int t = tid; t < 128; t += 64)
    __builtin_prefetch((const char*)rb + t * 128, 0, 3);

  if (tid < KNN) nbrS[tid] = idx[(size_t)q * KNN + tid];
  __syncthreads();

  if (tid < KNN) {
    const int j = nbrS[tid];
    const float* pq = xyz + (size_t)(b * N + n) * 3;
    const float* pj = xyz + (size_t)(b * N + j) * 3;
    relS[tid * 4 + 0] = pj[0] - pq[0];
    relS[tid * 4 + 1] = pj[1] - pq[1];
    relS[tid * 4 + 2] = pj[2] - pq[2];
  }
  // async gather of 24 neighbor feature rows (64 floats each), 16B chunks
  for (int t = tid; t < KNN * 16; t += 64) {
    const int k = t >> 4, seg = t & 15;
    async_copy_b128((unsigned)((k * DD + seg * 4) * 4),
                    feat + ((size_t)b * N + nbrS[k]) * DD + seg * 4);
  }
  async_wait0();
  __syncthreads();

  float Mx = -3.4e38f, S = 0.f, O = 0.f;
  const float* rw0 = rw + 0 * DD * DD + tid;
  const float* rw1 = rw + 1 * DD * DD + tid;
  const float* rw2 = rw + 2 * DD * DD + tid;
  const float* rbp = rb + tid;

  for (int k = 0; k < KNN; ++k) {
    const float rx = relS[k * 4 + 0], ry = relS[k * 4 + 1], rz = relS[k * 4 + 2];
#pragma unroll 8
    for (int d1 = 0; d1 < DD; ++d1) {
      const int o = d1 << 6;
      float w = rbp[o];
      w = fmaf(rx, rw0[o], w);
      w = fmaf(ry, rw1[o], w);
      w = fmaf(rz, rw2[o], w);
      const float g  = nf[k * DD + d1];
      const float nM = fmaxf(Mx, w);
      const float e0 = __expf(Mx - nM);       // rescale old accumulators
      const float e1 = __expf(w - nM);
      S  = fmaf(S, e0, e1);
      O  = fmaf(O, e0, g * e1);
      Mx = nM;
    }
  }
  out[(size_t)q * DD + tid] = O / S;
}

// second tuple output: the python int N, stored as float after the tensor
__global__ void tail_kernel(float* __restrict__ out, int off, float val) {
  if (threadIdx.x == 0 && blockIdx.x == 0) out[off] = val;
}

// ---------------------------------------------------------------------------
extern "C" void kernel_launch(void* const* d_in, const int* in_sizes, int n_in,
                              void* d_out, int out_size, void* d_ws, size_t ws_size,
                              hipStream_t stream) {
  const float* feature = (const float*)d_in[0];   // [B,N,64]
  const float* xyz     = (const float*)d_in[1];   // [B,N,3]
  const float* pw1     = (const float*)d_in[2];   // [64,64]
  const float* pb1     = (const float*)d_in[3];   // [64]
  const float* pw2     = (const float*)d_in[4];   // [64,64]
  const float* pb2     = (const float*)d_in[5];   // [64]
  const float* rw      = (const float*)d_in[6];   // [3,4096]
  const float* rb      = (const float*)d_in[7];   // [4096]
  const float* sw      = (const float*)d_in[8];   // [64,64]
  const float* sb      = (const float*)d_in[9];   // [64]

  const int BN = in_sizes[1] / 3;                 // B*N total points (2048)
  const int N  = NPTS;                            // points per batch (fixed)

  float* feat = (float*)d_ws;                     // [BN,64]
  float* agg  = feat + (size_t)BN * DD;           // [BN,64]
  int*   idx  = (int*)(agg + (size_t)BN * DD);    // [BN,24]

  // 1) prefix MLP (WMMA f32, weights via async-to-LDS)
  mlp2_kernel<<<BN / 64, 128, 0, stream>>>(feature, pw1, pb1, pw2, pb2, feat, BN);
  // 2) KNN selection
  knn_kernel<<<BN / 8, 256, 0, stream>>>(xyz, idx, N);
  // 3) fused weight-gen + softmax + aggregation (async neighbor gather)
  agg_kernel<<<BN, 64, 0, stream>>>(xyz, feat, idx, rw, rb, agg, N);
  // 4) final projection (WMMA f32) straight into d_out
  gemm_bias_kernel<<<BN / 64, 128, 0, stream>>>(agg, sw, sb, (float*)d_out, BN);
  // 5) tuple's second output
  if (out_size > BN * DD)
    tail_kernel<<<1, 1, 0, stream>>>((float*)d_out, BN * DD, (float)N);
}